// WeightedSAGEConv_55259049230672
// MI455X (gfx1250) — compile-verified
//
#include <hip/hip_runtime.h>

typedef __attribute__((ext_vector_type(2))) float v2f;
typedef __attribute__((ext_vector_type(8))) float v8f;

#define D_FEAT 128

// ---------------------------------------------------------------------------
// Phase 0: zero the accumulator workspace (acc[N*128] ++ denom[N], contiguous)
// ---------------------------------------------------------------------------
__global__ __launch_bounds__(256) void wsage_zero(float* __restrict__ p, long long n) {
  long long i = (long long)blockIdx.x * 256 + threadIdx.x;
  if (i < n) p[i] = 0.0f;
}

// ---------------------------------------------------------------------------
// Phase 1: wave-per-edge weighted scatter.
//   Each wave: coalesced 512B gather of x[col] (float4/lane), scale by w,
//   4x global_atomic_add_f32 per lane into acc[row], lane0 adds w to denom.
//   x + acc + denom all fit in the 192MB L2 -> L2-resident atomics.
// ---------------------------------------------------------------------------
__global__ __launch_bounds__(256) void wsage_scatter(
    const float* __restrict__ x,
    const long long* __restrict__ row_idx,   // destination
    const long long* __restrict__ col_idx,   // source
    const float* __restrict__ ew,
    float* __restrict__ acc,
    float* __restrict__ denom,
    int E) {
  int e    = (int)(((long long)blockIdx.x * 256 + threadIdx.x) >> 5);
  int lane = threadIdx.x & 31;
  if (e >= E) return;

  long long r = row_idx[e];
  long long c = col_idx[e];
  float w = ew[e];

  const float4 v = *(const float4*)(x + c * D_FEAT + lane * 4);
  float* dst = acc + r * D_FEAT + lane * 4;
  unsafeAtomicAdd(dst + 0, v.x * w);
  unsafeAtomicAdd(dst + 1, v.y * w);
  unsafeAtomicAdd(dst + 2, v.z * w);
  unsafeAtomicAdd(dst + 3, v.w * w);
  if (lane == 0) unsafeAtomicAdd(denom + r, w);
}

// ---------------------------------------------------------------------------
// Phase 2: fused normalize + dual GEMM via V_WMMA_F32_16X16X4_F32.
//   One wave owns a 16-row group and computes ALL 128 output columns with
//   8 independent v8f accumulators: A (x row-chunk + normalized acc chunk)
//   is loaded ONCE per K-step and reused across 8 column tiles x 2 weight
//   matrices = 16 WMMAs per K-step, two per accumulator (self then neigh)
//   (D->C RAW chain within each pair). Weights are L0/L2-resident.
//   out[m][n] = sum_k x[m][k]*Wself[n][k] + (acc[m][k]/denom[m])*Wneigh[n][k]
//             + bself[n] + bneigh[n]
// ---------------------------------------------------------------------------
__global__ __launch_bounds__(256) void wsage_gemm(
    const float* __restrict__ x,
    const float* __restrict__ acc,
    const float* __restrict__ denom,
    const float* __restrict__ Wself,
    const float* __restrict__ bself,
    const float* __restrict__ Wneigh,
    const float* __restrict__ bneigh,
    float* __restrict__ out,
    int N, int rowTiles) {
  const int wave = threadIdx.x >> 5;
  const int lane = threadIdx.x & 31;
  const int rowTile = blockIdx.x * 8 + wave;
  if (rowTile >= rowTiles) return;       // wave-uniform exit

  const int row0 = rowTile * 16;
  const int m    = lane & 15;            // A row within tile / B col within tile
  const int half = lane >> 4;
  const int kh   = half * 2;             // lanes 0-15: k,k+1 ; lanes 16-31: k+2,k+3

  int arow = row0 + m;
  if (arow > N - 1) arow = N - 1;        // clamp loads; stores handled below

  const float inv = 1.0f / fmaxf(denom[arow], 1e-12f);
  const float* xp  = x      + (long long)arow * D_FEAT + kh;
  const float* np  = acc    + (long long)arow * D_FEAT + kh;
  const float* wsp = Wself  + m * D_FEAT + kh;   // + t*16*128 per col tile
  const float* wnp = Wneigh + m * D_FEAT + kh;

  v8f cacc[8];
#pragma unroll
  for (int t = 0; t < 8; ++t) cacc[t] = (v8f){0.f,0.f,0.f,0.f,0.f,0.f,0.f,0.f};

  for (int k = 0; k < D_FEAT; k += 4) {
    float2 ax = *(const float2*)(xp + k);
    float2 an = *(const float2*)(np + k);
    v2f A0 = { ax.x,       ax.y       };
    v2f A1 = { an.x * inv, an.y * inv };
#pragma unroll
    for (int t = 0; t < 8; ++t) {
      float2 bs = *(const float2*)(wsp + t * 16 * D_FEAT + k);
      float2 bn = *(const float2*)(wnp + t * 16 * D_FEAT + k);
      v2f B0 = { bs.x, bs.y };
      v2f B1 = { bn.x, bn.y };
      cacc[t] = __builtin_amdgcn_wmma_f32_16x16x4_f32(false, A0, false, B0,
                                                      (short)0, cacc[t], false, false);
      cacc[t] = __builtin_amdgcn_wmma_f32_16x16x4_f32(false, A1, false, B1,
                                                      (short)0, cacc[t], false, false);
    }
  }

  // D layout: VGPR j -> row j (lanes 0-15) / row j+8 (lanes 16-31), N = lane&15
  const int rbase = row0 + half * 8;
  if (row0 + 16 <= N) {
    // Full tile (common case): straight-line stores, no per-lane guards.
#pragma unroll
    for (int t = 0; t < 8; ++t) {
      const int n = t * 16 + m;
      const float bias = bself[n] + bneigh[n];
#pragma unroll
      for (int j = 0; j < 8; ++j) {
        out[(long long)(rbase + j) * D_FEAT + n] = cacc[t][j] + bias;
      }
    }
  } else {
#pragma unroll
    for (int t = 0; t < 8; ++t) {
      const int n = t * 16 + m;
      const float bias = bself[n] + bneigh[n];
#pragma unroll
      for (int j = 0; j < 8; ++j) {
        int r = rbase + j;
        if (r < N) out[(long long)r * D_FEAT + n] = cacc[t][j] + bias;
      }
    }
  }
}

// ---------------------------------------------------------------------------
extern "C" void kernel_launch(void* const* d_in, const int* in_sizes, int n_in,
                              void* d_out, int out_size, void* d_ws, size_t ws_size,
                              hipStream_t stream) {
  const float*     x      = (const float*)d_in[0];
  const long long* ei     = (const long long*)d_in[1];  // [2, E] int64
  const float*     ew     = (const float*)d_in[2];
  const float*     Wself  = (const float*)d_in[3];
  const float*     bself  = (const float*)d_in[4];
  const float*     Wneigh = (const float*)d_in[5];
  const float*     bneigh = (const float*)d_in[6];
  float*           out    = (float*)d_out;

  const int N = in_sizes[0] / D_FEAT;
  const int E = in_sizes[2];

  // Workspace: acc[N*128] then denom[N] (fp32), ~51.6 MB total.
  float* acc   = (float*)d_ws;
  float* denom = acc + (long long)N * D_FEAT;

  long long ztotal = (long long)N * D_FEAT + N;
  int zblocks = (int)((ztotal + 255) / 256);
  wsage_zero<<<zblocks, 256, 0, stream>>>(acc, ztotal);

  int sblocks = (E + 7) / 8;  // 8 waves (edges) per 256-thread block
  wsage_scatter<<<sblocks, 256, 0, stream>>>(x, ei, ei + E, ew, acc, denom, E);

  int rowTiles = (N + 15) / 16;
  int gblocks  = (rowTiles + 7) / 8;   // 8 row tiles (waves) per block
  wsage_gemm<<<gblocks, 256, 0, stream>>>(x, acc, denom, Wself, bself,
                                          Wneigh, bneigh, out, N, rowTiles);
}